// Construct_Graph_59880434041330
// MI455X (gfx1250) — compile-verified
//
#include <hip/hip_runtime.h>
#include <hip/hip_bf16.h>

// KNN graph construction for MI455X (gfx1250, wave32).
// Memory-bound problem: 512MB of output stores dominate (~22us at 23.3TB/s);
// the 34-GFLOP gram matrix uses exact-fp32 V_WMMA_F32_16X16X4_F32 so the
// top-15 ordering matches the f32 reference. X (8MB) is L2-resident (192MB
// L2), so re-streaming all columns per row-band is effectively free.
//
// Round-2 change: WMMA operands are fetched at lane-dependent addresses that
// match the ISA's A/B VGPR striping directly (ds_load_b64 for A, two dword
// loads for B), eliminating the 6 v_cndmask per WMMA the previous version
// spent unpacking b128 loads.

#define N     8192
#define F     256
#define KNN   15
#define CHUNK 128          // columns processed per workgroup iteration (8 tiles)
#define SPAD  (CHUNK + 1)  // LDS row padding to stagger banks for the top-k scan

typedef __attribute__((ext_vector_type(2))) float v2f;
typedef __attribute__((ext_vector_type(8))) float v8f;

// ---------------------------------------------------------------- row norms
__global__ void __launch_bounds__(256) row_norms_kernel(
    const float* __restrict__ x, float* __restrict__ sq) {
  const int wave = threadIdx.x >> 5;
  const int lane = threadIdx.x & 31;
  const int row  = blockIdx.x * 8 + wave;
  const float4* xr = (const float4*)(x + (size_t)row * F);
  float s = 0.f;
  #pragma unroll
  for (int i = lane; i < F / 4; i += 32) {
    float4 v = xr[i];
    s += v.x * v.x + v.y * v.y + v.z * v.z + v.w * v.w;
  }
  #pragma unroll
  for (int off = 16; off > 0; off >>= 1) s += __shfl_xor(s, off, 32);
  if (lane == 0) sq[row] = s;
}

// ---------------------------------------------------------------- zero adj
__global__ void zero_kernel(float4* __restrict__ p, long n4) {
  long idx    = (long)blockIdx.x * blockDim.x + threadIdx.x;
  long stride = (long)gridDim.x * blockDim.x;
  float4 z = {0.f, 0.f, 0.f, 0.f};
  for (long i = idx; i < n4; i += stride) p[i] = z;
}

// ------------------------------------------------ fused WMMA score + top-k
// One workgroup (8 waves) owns 16 rows. Per 128-column chunk each wave builds
// one 16x16 score tile with V_WMMA_F32_16X16X4_F32, then 16 threads merge the
// chunk into their exact per-row top-15.
__global__ void __launch_bounds__(256) score_topk_kernel(
    const float* __restrict__ x, const float* __restrict__ sq,
    int* __restrict__ topk) {
  __shared__ float Alds[16 * F];      // 16KB: this band's 16 rows of X
  __shared__ float Slds[16 * SPAD];   // padded 16x128 score slab

  const int rowbase = blockIdx.x * 16;
  const int tid  = threadIdx.x;
  const int wave = tid >> 5;
  const int lane = tid & 31;
  const int hi   = lane >> 4;   // which half-wave
  const int ln   = lane & 15;

  // Stage A band into LDS (float4, fully coalesced).
  for (int i = tid; i < 16 * F / 4; i += 256)
    ((float4*)Alds)[i] = ((const float4*)(x + (size_t)rowbase * F))[i];
  __syncthreads();

  // Per-row top-15 state (threads 0..15 own row rowbase+tid).
  float best_v[KNN];
  int   best_i[KNN];
  #pragma unroll
  for (int q = 0; q < KNN; ++q) { best_v[q] = -__builtin_inff(); best_i[q] = 0; }

  // A 16x4 layout: lanes 0-15 hold M=ln with K = {4kk+0, 4kk+1}; lanes 16-31
  // hold K = {4kk+2, 4kk+3}. That is a CONTIGUOUS pair at element offset
  // ln*F + 4kk + 2*hi -> one ds_load_b64 per k-step, no selects.
  const v2f* arow = (const v2f*)(Alds + ln * F + 2 * hi);

  for (int chunk = 0; chunk < N / CHUNK; ++chunk) {
    const int colbase = chunk * CHUNK + wave * 16;
    const int col     = colbase + ln;                 // this lane's B column
    // B 4x16 layout: VGPR0 = row K=hi, VGPR1 = row K=hi+2 (N = ln striped
    // across the half-wave). Per k-step the lane needs X[col][4kk+hi] and
    // X[col][4kk+hi+2]: two dword loads at a lane-dependent base.
    const float* bp  = x + (size_t)col * F + hi;
    const float sqj = sq[col];

    v8f c = {};
    for (int kk = 0; kk < F / 4; ++kk) {
      v2f a = arow[2 * kk];       // ds_load_b64: A pair for this half-wave
      v2f b;
      b.x = bp[4 * kk];           // K = hi   row of the 4x16 B tile
      b.y = bp[4 * kk + 2];       // K = hi+2 row
      c = __builtin_amdgcn_wmma_f32_16x16x4_f32(
          /*neg_a=*/false, a, /*neg_b=*/false, b,
          /*c_mod=*/(short)0, c, /*reuse_a=*/false, /*reuse_b=*/false);
    }

    // C/D layout: VGPR r -> M = r + 8*hi, N = ln. Monotone score 2g - sq_j,
    // diagonal masked to -inf (matches reference's -inf on S's diagonal).
    #pragma unroll
    for (int r = 0; r < 8; ++r) {
      const int m = r + 8 * hi;
      float val = 2.f * c[r] - sqj;
      if (rowbase + m == col) val = -__builtin_inff();
      Slds[m * SPAD + wave * 16 + ln] = val;
    }
    __syncthreads();

    if (tid < 16) {   // exact insertion into sorted top-15
      const float* srow = Slds + tid * SPAD;
      for (int n = 0; n < CHUNK; ++n) {
        float v = srow[n];
        if (v > best_v[KNN - 1]) {
          int pos = KNN - 1;
          while (pos > 0 && best_v[pos - 1] < v) {
            best_v[pos] = best_v[pos - 1];
            best_i[pos] = best_i[pos - 1];
            --pos;
          }
          best_v[pos] = v;
          best_i[pos] = chunk * CHUNK + n;
        }
      }
    }
    __syncthreads();
  }

  if (tid < 16) {
    #pragma unroll
    for (int q = 0; q < KNN; ++q)
      topk[(size_t)(rowbase + tid) * KNN + q] = best_i[q];
  }
}

// ------------------------------------------------------- symmetric scatter
__global__ void scatter_kernel(const int* __restrict__ topk,
                               float* __restrict__ adj) {
  int idx = blockIdx.x * blockDim.x + threadIdx.x;
  if (idx >= N * KNN) return;
  int i = idx / KNN;
  int j = topk[idx];
  adj[(size_t)i * N + j] = 1.0f;   // racing identical stores: deterministic
  adj[(size_t)j * N + i] = 1.0f;
}

// -------------------------------------------------------- row normalization
__global__ void __launch_bounds__(256) normalize_kernel(
    const float* __restrict__ adj, float* __restrict__ ahat) {
  __shared__ float red[8];
  const int row  = blockIdx.x;
  const int wave = threadIdx.x >> 5;
  const int lane = threadIdx.x & 31;
  const float4* a4 = (const float4*)(adj + (size_t)row * N);
  float4*       o4 = (float4*)(ahat + (size_t)row * N);

  float s = 0.f;
  for (int i = threadIdx.x; i < N / 4; i += 256) {
    float4 v = a4[i];
    s += v.x + v.y + v.z + v.w;
  }
  #pragma unroll
  for (int off = 16; off > 0; off >>= 1) s += __shfl_xor(s, off, 32);
  if (lane == 0) red[wave] = s;
  __syncthreads();
  if (threadIdx.x == 0) {
    float t = 0.f;
    #pragma unroll
    for (int w = 0; w < 8; ++w) t += red[w];
    red[0] = (t > 0.f) ? (1.0f / t) : 0.0f;   // inf -> 0 as in reference
  }
  __syncthreads();
  const float inv = red[0];
  for (int i = threadIdx.x; i < N / 4; i += 256) {
    float4 v = a4[i];
    float4 o = {v.x * inv, v.y * inv, v.z * inv, v.w * inv};
    o4[i] = o;
  }
}

// ------------------------------------------------------------------ launch
extern "C" void kernel_launch(void* const* d_in, const int* in_sizes, int n_in,
                              void* d_out, int out_size, void* d_ws, size_t ws_size,
                              hipStream_t stream) {
  const float* x = (const float*)d_in[0];
  float* adj  = (float*)d_out;                    // output 0: adjacency [N,N]
  float* ahat = adj + (size_t)N * N;              // output 1: A_hat     [N,N]

  float* sq   = (float*)d_ws;                     // N floats
  int*   topk = (int*)((char*)d_ws + (size_t)N * sizeof(float));  // N*KNN ints

  row_norms_kernel<<<N / 8, 256, 0, stream>>>(x, sq);
  zero_kernel<<<2048, 256, 0, stream>>>((float4*)adj, (long)N * N / 4);
  score_topk_kernel<<<N / 16, 256, 0, stream>>>(x, sq, topk);
  scatter_kernel<<<(N * KNN + 255) / 256, 256, 0, stream>>>(topk, adj);
  normalize_kernel<<<N, 256, 0, stream>>>(adj, ahat);
}